// SVD_pp_86500641342004
// MI455X (gfx1250) — compile-verified
//
#include <hip/hip_runtime.h>

// SVD++ scoring for MI455X (gfx1250, wave32).
// Constants from reference: B=16384, R=2,000,000, U=200,000, E=128.

#define EMB 128
#define SENTINEL 0x7FFFFFFF

typedef __attribute__((ext_vector_type(2))) float v2f;
typedef __attribute__((ext_vector_type(8))) float v8f;

// ---------------------------------------------------------------- init slots/counts
__global__ __launch_bounds__(256)
void init_slots_kernel(int* __restrict__ slot, float* __restrict__ counts, int U) {
    int i = blockIdx.x * blockDim.x + threadIdx.x;
    if (i < U) { slot[i] = SENTINEL; counts[i] = 0.0f; }
}

// ---------------------------------------------------------------- zero accumulator
__global__ __launch_bounds__(256)
void zero_acc_kernel(float4* __restrict__ acc4, int n4) {
    int i = blockIdx.x * blockDim.x + threadIdx.x;
    if (i < n4) { float4 z; z.x = 0.f; z.y = 0.f; z.z = 0.f; z.w = 0.f; acc4[i] = z; }
}

// ---------------------------------------------------------------- mark batch users
__global__ __launch_bounds__(256)
void mark_kernel(const int* __restrict__ u, int* __restrict__ slot, int B) {
    int i = blockIdx.x * blockDim.x + threadIdx.x;
    if (i < B) atomicMin(&slot[u[i]], i);
}

// ---------------------------------------------------------------- masked segment sum
// One wave32 per rating. Skip ratings whose user is not in the batch (~92%).
// Each lane handles 4 consecutive floats (32 lanes * 4 = 128 = EMB).
__global__ __launch_bounds__(256)
void seg_accum_kernel(const int* __restrict__ rating_users,
                      const int* __restrict__ rating_items,
                      const float* __restrict__ item_implicit_emb,
                      const int* __restrict__ slot,
                      float* __restrict__ counts,
                      float* __restrict__ acc,
                      int R) {
    int wave = (blockIdx.x * blockDim.x + threadIdx.x) >> 5;
    int lane = threadIdx.x & 31;
    if (wave >= R) return;
    int user = rating_users[wave];          // uniform across the wave
    int s = slot[user];
    if (s == SENTINEL) return;              // user not in batch: drop
    int item = rating_items[wave];
    const float4 val =
        ((const float4*)(item_implicit_emb + (size_t)item * EMB))[lane];
    float* dst = acc + (size_t)s * EMB + lane * 4;
    atomicAdd(dst + 0, val.x);
    atomicAdd(dst + 1, val.y);
    atomicAdd(dst + 2, val.z);
    atomicAdd(dst + 3, val.w);
    if (lane == 0) atomicAdd(&counts[user], 1.0f);
}

// ---------------------------------------------------------------- WMMA scoring
// One wave32 per 16 batch rows. D = A x B^T via v_wmma_f32_16x16x4_f32,
// A[m][k] = item_emb[v[m]][k], B^T so B[k][n] = X[n][k], X = user_emb + scale*acc.
// Diagonal of the 16x16 f32 tile = the 16 dot products.
__global__ __launch_bounds__(32)
void score_kernel(const int* __restrict__ u, const int* __restrict__ v,
                  const float* __restrict__ user_emb,
                  const float* __restrict__ user_emb_bias,
                  const float* __restrict__ item_emb,
                  const float* __restrict__ item_emb_bias,
                  const float* __restrict__ mean,
                  const int* __restrict__ slot,
                  const float* __restrict__ counts,
                  const float* __restrict__ acc,
                  float* __restrict__ out) {
    __shared__ float X[16 * EMB];     // 8 KB: X rows (user + scaled implicit)
    __shared__ float Dtile[16 * 16];  // 1 KB: full WMMA result tile
    __shared__ float biasLds[16];

    const int lane = threadIdx.x;     // 0..31
    const int base = blockIdx.x * 16; // batch rows [base, base+16)

    // ---- Phase 1: build X = user_emb[u] + |N(u)|^-1/2 * implicit_sum, into LDS.
    // 512 float4 stores spread over 32 lanes * 16 iterations.
    #pragma unroll
    for (int i = 0; i < 16; ++i) {
        int linear4 = i * 32 + lane;      // float4 index in [0,512)
        int row = linear4 >> 5;           // 32 float4 per 128-float row
        int col4 = linear4 & 31;
        int uid = u[base + row];
        int s = slot[uid];                // always valid: set by mark_kernel
        float cnt = counts[uid];
        float scale = (cnt > 0.0f) ? rsqrtf(cnt) : 0.0f;
        float4 ue = ((const float4*)(user_emb + (size_t)uid * EMB))[col4];
        float4 im = ((const float4*)(acc + (size_t)s * EMB))[col4];
        float4 x;
        x.x = ue.x + scale * im.x;
        x.y = ue.y + scale * im.y;
        x.z = ue.z + scale * im.z;
        x.w = ue.w + scale * im.w;
        ((float4*)X)[linear4] = x;
    }
    if (lane < 16) {
        int uid = u[base + lane];
        int vid = v[base + lane];
        biasLds[lane] = user_emb_bias[uid] + item_emb_bias[vid] + mean[0];
    }
    __syncthreads();

    // ---- Phase 2: 32 chained f32 WMMAs over K=128.
    // A 16x4 f32 layout: lanes 0-15 hold (M=lane, K={k0,k0+1}) in vgpr{0,1};
    // lanes 16-31 hold (M=lane-16, K={k0+2,k0+3}).
    // B 4x16 layout mirrors C/D row striping: vgpr0 = rows {k0 | k0+2},
    // vgpr1 = rows {k0+1 | k0+3}, N = lane%16.  B[k][n] = X[n][k].
    const int m = lane & 15;
    const int half = lane >> 4;                   // 0 or 1
    const float* Irow = item_emb + (size_t)v[base + m] * EMB;
    const float* Xrow = X + m * EMB;
    __builtin_prefetch(Irow, 0, 0);               // global_prefetch_b8

    v8f c = {};
    #pragma unroll
    for (int kk = 0; kk < 32; ++kk) {
        int k = kk * 4 + half * 2;
        v2f a; a.x = Irow[k]; a.y = Irow[k + 1];
        v2f b; b.x = Xrow[k]; b.y = Xrow[k + 1];
        c = __builtin_amdgcn_wmma_f32_16x16x4_f32(
                /*neg_a=*/false, a, /*neg_b=*/false, b,
                /*c_mod=*/(short)0, c, /*reuse_a=*/false, /*reuse_b=*/false);
    }

    // ---- Phase 3: spill D tile to LDS (vgpr g -> row g + 8*half, col m),
    // then lanes 0..15 read the diagonal and add biases.
    #pragma unroll
    for (int g = 0; g < 8; ++g) {
        Dtile[(g + half * 8) * 16 + m] = c[g];
    }
    __syncthreads();
    if (lane < 16) {
        out[base + lane] = Dtile[lane * 16 + lane] + biasLds[lane];
    }
}

// ---------------------------------------------------------------- launch
extern "C" void kernel_launch(void* const* d_in, const int* in_sizes, int n_in,
                              void* d_out, int out_size, void* d_ws, size_t ws_size,
                              hipStream_t stream) {
    const int*   u                 = (const int*)  d_in[0];
    const int*   v                 = (const int*)  d_in[1];
    const int*   rating_users      = (const int*)  d_in[2];
    const int*   rating_items      = (const int*)  d_in[3];
    const float* user_emb          = (const float*)d_in[4];
    const float* user_emb_bias     = (const float*)d_in[5];
    const float* item_emb          = (const float*)d_in[6];
    const float* item_emb_bias     = (const float*)d_in[7];
    const float* item_implicit_emb = (const float*)d_in[8];
    const float* mean              = (const float*)d_in[9];
    float* out = (float*)d_out;

    const int B = in_sizes[0];              // 16384
    const int R = in_sizes[2];              // 2,000,000
    const int U = in_sizes[5];              // 200,000 (user_emb_bias is [U,1])

    // Workspace layout (256B-aligned regions): slot[U] int, counts[U] float,
    // acc[B*EMB] float  -> ~10 MB total.
    char* ws = (char*)d_ws;
    size_t off = 0;
    int* slot = (int*)(ws + off);
    off += ((size_t)U * sizeof(int) + 255) & ~(size_t)255;
    float* counts = (float*)(ws + off);
    off += ((size_t)U * sizeof(float) + 255) & ~(size_t)255;
    float* acc = (float*)(ws + off);

    // 1) init slot/count tables
    init_slots_kernel<<<(U + 255) / 256, 256, 0, stream>>>(slot, counts, U);
    // 2) zero the per-representative accumulator
    int n4 = B * (EMB / 4);
    zero_acc_kernel<<<(n4 + 255) / 256, 256, 0, stream>>>((float4*)acc, n4);
    // 3) flag batch users with their representative batch row
    mark_kernel<<<(B + 255) / 256, 256, 0, stream>>>(u, slot, B);
    // 4) masked segment-sum: one wave per rating (8 waves / 256-thread block)
    seg_accum_kernel<<<(R + 7) / 8, 256, 0, stream>>>(
        rating_users, rating_items, item_implicit_emb, slot, counts, acc, R);
    // 5) WMMA scoring: one wave per 16 batch rows
    score_kernel<<<B / 16, 32, 0, stream>>>(
        u, v, user_emb, user_emb_bias, item_emb, item_emb_bias, mean,
        slot, counts, acc, out);
}